// BeamOperator_10453950398865
// MI455X (gfx1250) — compile-verified
//
#include <hip/hip_runtime.h>

// ---------------------------------------------------------------------------
// Beam energy reduction for MI455X (gfx1250, wave32).
//   Pass 1: per-element von-Karman energy density, quadrature-summed, reduced
//           per-block (shfl wave32 reduce + LDS) -> d_ws partials.
//   Pass 2: one wave folds the 2048 block partials with V_WMMA_F32_16X16X4_F32
//           (B = ones -> row-sum accumulate in C), then writes the scalar.
// Bandwidth-bound: ~48 MB unique traffic -> ~2 us floor at 23.3 TB/s.
// Deterministic: fixed grid, fixed reduction order, no float atomics.
// ---------------------------------------------------------------------------

#define TPB   256
#define NBLK  2048   // multiple of 64 so the WMMA reduction needs no tail

typedef __attribute__((ext_vector_type(2))) float v2f;
typedef __attribute__((ext_vector_type(8))) float v8f;

__global__ __launch_bounds__(TPB) void beam_partials_kernel(
    const float* __restrict__ nv,      // (N_NODES, 3) nodal values
    const float* __restrict__ xs,      // (N_NODES,)   coords
    const int*   __restrict__ elems,   // (N_ELEM, 2)  connectivity (int32)
    float*       __restrict__ partials,
    int n_elem)
{
    const float hEA = 0.5f * 1000.0f;  // 0.5*EA
    const float hEI = 0.5f * 10.0f;    // 0.5*EI
    // Gauss points / weights (3-pt)
    const float XI[3] = { -0.774596669241483f, 0.0f, 0.774596669241483f };
    const float WQ[3] = { 5.0f/9.0f, 8.0f/9.0f, 5.0f/9.0f };

    float acc = 0.0f;
    const int tid    = blockIdx.x * blockDim.x + threadIdx.x;
    const int stride = gridDim.x * blockDim.x;

    for (int e = tid; e < n_elem; e += stride) {
        // hint-prefetch next chunk's nodal data (global_prefetch_b8)
        int en = e + stride;
        if (en < n_elem) __builtin_prefetch(nv + 3 * (size_t)en, 0, 1);

        const int n0 = elems[2 * e];
        const int n1 = elems[2 * e + 1];
        const float* p0 = nv + 3 * (size_t)n0;
        const float* p1 = nv + 3 * (size_t)n1;
        const float u1 = p0[0], w1 = p0[1], t1 = p0[2];
        const float u2 = p1[0], w2 = p1[1], t2 = p1[2];

        const float L     = xs[n1] - xs[n0];
        const float invL  = 1.0f / L;
        const float invJ  = 2.0f * invL;        // d(xi)/dx
        const float invJ2 = invJ * invJ;
        const float detJ  = 0.5f * L;
        const float du    = (u2 - u1) * invL;   // du/dx, constant over xi
        const float dwm   = w1 - w2;            // dH3 = -dH1, ddH3 = -ddH1

        #pragma unroll
        for (int q = 0; q < 3; ++q) {
            const float xi  = XI[q];
            const float xi2 = xi * xi;
            // factored Hermite derivatives (L pulled out of theta terms)
            const float cA = (3.0f*xi2 - 3.0f) * 0.25f;            // dH1
            const float c2 = (3.0f*xi2 - 2.0f*xi - 1.0f) * 0.125f; // dH2/L
            const float c4 = (3.0f*xi2 + 2.0f*xi - 1.0f) * 0.125f; // dH4/L
            const float dB = 1.5f * xi;                            // ddH1
            const float d2 = (6.0f*xi - 2.0f) * 0.125f;            // ddH2/L
            const float d4 = (6.0f*xi + 2.0f) * 0.125f;            // ddH4/L

            const float dw_dxi   = fmaf(cA, dwm, L * fmaf(c2, t1, c4 * t2));
            const float d2w_dxi2 = fmaf(dB, dwm, L * fmaf(d2, t1, d4 * t2));
            const float dw_dx    = dw_dxi * invJ;
            const float d2w_dx2  = d2w_dxi2 * invJ2;

            const float eps = fmaf(0.5f * dw_dx, dw_dx, du);
            const float psi = fmaf(hEA * eps, eps, hEI * d2w_dx2 * d2w_dx2);
            acc = fmaf(psi, detJ * WQ[q], acc);
        }
    }

    // wave32 reduce
    #pragma unroll
    for (int off = 16; off > 0; off >>= 1)
        acc += __shfl_down(acc, off, 32);

    __shared__ float sm[TPB / 32];
    if ((threadIdx.x & 31) == 0) sm[threadIdx.x >> 5] = acc;
    __syncthreads();
    if (threadIdx.x == 0) {
        float s = 0.0f;
        #pragma unroll
        for (int i = 0; i < TPB / 32; ++i) s += sm[i];
        partials[blockIdx.x] = s;
    }
}

// Single-wave final reduction. n must be a multiple of 64.
__global__ __launch_bounds__(32) void beam_reduce_kernel(
    const float* __restrict__ partials,
    float*       __restrict__ out,
    int n)
{
    const int lane = threadIdx.x;  // 32 threads, EXEC all ones (WMMA req.)
#if defined(__gfx1250__) && __has_builtin(__builtin_amdgcn_wmma_f32_16x16x4_f32)
    // D = A(16x4) * B(4x16) + C ; B = ones => D[m][n] = sum_k A[m][k] + C[m][n]
    v8f c = {};
    v2f b; b.x = 1.0f; b.y = 1.0f;
    for (int base = 0; base < n; base += 64) {
        v2f a;
        a.x = partials[base + lane];
        a.y = partials[base + 32 + lane];
        c = __builtin_amdgcn_wmma_f32_16x16x4_f32(
            /*neg_a=*/false, a, /*neg_b=*/false, b,
            /*c_mod=*/(short)0, c, /*reuse_a=*/false, /*reuse_b=*/false);
    }
    // C layout (32-bit 16x16): lanes 0-15 VGPR j -> M=j, N=lane;
    //                          lanes 16-31 VGPR j -> M=j+8, N=lane-16.
    // Column 0 lives in lane 0 (rows 0-7) and lane 16 (rows 8-15).
    float s = c[0] + c[1] + c[2] + c[3] + c[4] + c[5] + c[6] + c[7];
    const float s_hi = __shfl(s, 16, 32);
    if (lane == 0) out[0] = s + s_hi;
#else
    float s = 0.0f;
    for (int i = lane; i < n; i += 32) s += partials[i];
    #pragma unroll
    for (int off = 16; off > 0; off >>= 1)
        s += __shfl_down(s, off, 32);
    if (lane == 0) out[0] = s;
#endif
}

extern "C" void kernel_launch(void* const* d_in, const int* in_sizes, int n_in,
                              void* d_out, int out_size, void* d_ws, size_t ws_size,
                              hipStream_t stream) {
    const float* nv    = (const float*)d_in[0];   // nodal_values (N,3) f32
    const float* xs    = (const float*)d_in[1];   // coords (N,)      f32
    const int*   elems = (const int*)d_in[2];     // elements (N-1,2) int32
    float*       out   = (float*)d_out;           // scalar f32
    float*       parts = (float*)d_ws;            // NBLK f32 partials

    const int n_elem = in_sizes[2] / 2;

    beam_partials_kernel<<<NBLK, TPB, 0, stream>>>(nv, xs, elems, parts, n_elem);
    beam_reduce_kernel<<<1, 32, 0, stream>>>(parts, out, NBLK);
}